// UnaryGatesUnitary_33466385170916
// MI455X (gfx1250) — compile-verified
//
#include <hip/hip_runtime.h>
#include <math.h>

// CDNA5 / gfx1250, wave32.
// Output: (4, 1, 4096, 4096, 1) complex64 = 512 MB. Pure store-BW bound
// (~22us at 23.3 TB/s). Values generated in-register via rank-1 WMMA
// outer products: out[(a,c),(b,d)] = K1[a,b] * K2[c,d].
// v3: fast native __sincosf instead of precise OCML sinf/cosf -- cuts the
// per-wave VALU/SALU count roughly in half so instruction issue cannot
// throttle the 512 MB store stream.

typedef float v2f __attribute__((ext_vector_type(2)));
typedef float v8f __attribute__((ext_vector_type(8)));

__global__ __launch_bounds__(256)
void kron_fill_wmma(const float* __restrict__ thetas, v2f* __restrict__ out) {
    const int lane   = threadIdx.x & 31;
    const int lane16 = lane & 15;
    const int hi     = lane >> 4;                       // 0: lanes 0-15, 1: lanes 16-31

    // one wave per output row; 4 groups * 4096 rows = 16384 waves total
    const int waveId = blockIdx.x * (blockDim.x >> 5) + (threadIdx.x >> 5);
    const int g    = waveId >> 12;       // group 0..3
    const int rrow = waveId & 4095;      // row within group
    const int a    = rrow >> 6;          // high 6 bits of row -> gates 0..5
    const int c    = rrow & 63;          // low 6 bits of row  -> gates 6..11

    // cos/sin of half-angles for this group's 12 gates (fast native trig;
    // |theta/2| is O(1) for randn angles, ~1 ulp vs precise libm)
    float ct[12], st[12];
#pragma unroll
    for (int m = 0; m < 12; ++m) {
        const float t = 0.5f * thetas[g * 12 + m];
        __sincosf(t, &st[m], &ct[m]);
    }

    // Row factors: k1v[t] = K1[a, t*16 + lane16], k2v[t] = K2[c, t*16 + lane16]
    // Gate entry G[r,s]: r==s -> cos ; (0,1) -> -sin ; (1,0) -> +sin.
    float k1v[4], k2v[4];
#pragma unroll
    for (int t = 0; t < 4; ++t) {
        const int col = t * 16 + lane16;
        float v1 = 1.0f, v2 = 1.0f;
#pragma unroll
        for (int m = 0; m < 6; ++m) {
            const int sh = 5 - m;
            // K1: row bits from a, gates m
            {
                const int rb = (a   >> sh) & 1;
                const int cb = (col >> sh) & 1;
                const float f = (rb == cb) ? ct[m] : (rb ? st[m] : -st[m]);
                v1 *= f;
            }
            // K2: row bits from c, gates 6+m
            {
                const int rb = (c   >> sh) & 1;
                const int cb = (col >> sh) & 1;
                const float f = (rb == cb) ? ct[6 + m] : (rb ? st[6 + m] : -st[6 + m]);
                v2 *= f;
            }
        }
        k1v[t] = v1;
        k2v[t] = v2;
    }

    const size_t rowbase = (size_t)(g * 4096 + rrow) * 4096u;

    // 16 rank-1 WMMA tiles cover the full 4096-wide output row.
    // A (16x4): column K=0 holds K1[a, b0*16 + m]; other K-slots zero.
    //   VGPR0: lanes 0-15 = A[m,0]  (lanes 16-31 = A[m,2] -> 0), VGPR1 (K=1,3) = 0.
    // B (4x16): row K=0 holds K2[c, d0*16 + n]; other rows zero (same striping).
#pragma unroll
    for (int b0 = 0; b0 < 4; ++b0) {
        v2f A;
        A.x = hi ? 0.0f : k1v[b0];
        A.y = 0.0f;
#pragma unroll
        for (int d0 = 0; d0 < 4; ++d0) {
            v2f B;
            B.x = hi ? 0.0f : k2v[d0];
            B.y = 0.0f;
            v8f Cz = {};
            // D[m,n] = K1[a, b0*16+m] * K2[c, d0*16+n]
            v8f D = __builtin_amdgcn_wmma_f32_16x16x4_f32(
                false, A, false, B, (short)0, Cz, false, false);

            // D layout: lane n<16 holds D[v, n] in VGPR v; lanes 16-31 hold D[v+8, n-16].
#pragma unroll
            for (int v = 0; v < 8; ++v) {
                const int m = v + 8 * hi;                 // b-offset within tile
                const int b = b0 * 16 + m;
                const int d = d0 * 16 + lane16;
                v2f val;
                val.x = D[v];                             // real part (kron is real)
                val.y = 0.0f;                             // imag part = 0
                __builtin_nontemporal_store(val, &out[rowbase + (size_t)(b * 64 + d)]);
            }
        }
    }
}

extern "C" void kernel_launch(void* const* d_in, const int* in_sizes, int n_in,
                              void* d_out, int out_size, void* d_ws, size_t ws_size,
                              hipStream_t stream) {
    const float* thetas = (const float*)d_in[0];   // 48 angles, float32
    v2f* out = (v2f*)d_out;                        // complex64 as interleaved {re, im}

    // 16384 output rows, one wave32 per row, 8 waves per 256-thread block
    dim3 grid(2048), block(256);
    kron_fill_wmma<<<grid, block, 0, stream>>>(thetas, out);
}